// LinearDecoder_55774445305978
// MI455X (gfx1250) — compile-verified
//
#include <hip/hip_runtime.h>
#include <hip/hip_bf16.h>

typedef float v2f __attribute__((ext_vector_type(2)));
typedef float v8f __attribute__((ext_vector_type(8)));

static constexpr int Tn     = 16384;
static constexpr int Hn     = 2048;
static constexpr int Kn     = 32;
static constexpr int START  = 30;
static constexpr int STOP   = 31;
static constexpr float NEGV = -10000.0f;
static constexpr int NCHUNK = 512;
static constexpr int CLEN   = Tn / NCHUNK;   // 32

// ---------------------------------------------------------------------------
// Kernel 1: feats[T,K] = hidden[T,H] @ W[K,H]^T via V_WMMA_F32_16X16X4_F32.
// One wave per 16-row tile; both 16-col tiles accumulated in the same wave so
// `hidden` is streamed exactly once (memory-bound: ~134 MB @ 23.3 TB/s).
// A-frag (16x4 f32): lanes 0-15 hold K={0,1}, lanes 16-31 hold K={2,3} -> one
// contiguous float2 per lane. B-frag (4x16) mirrors that layout over columns.
// ---------------------------------------------------------------------------
__global__ __launch_bounds__(256)
void feats_gemm_wmma(const float* __restrict__ hidden,
                     const float* __restrict__ W,
                     float* __restrict__ feats) {
    const int wave = threadIdx.x >> 5;
    const int lane = threadIdx.x & 31;
    const int tile = blockIdx.x * (blockDim.x >> 5) + wave;   // 0..1023
    const int row0 = tile * 16;
    const int half = lane >> 4;          // 0 -> K pair (0,1); 1 -> K pair (2,3)
    const int l15  = lane & 15;

    const float* aRow  = hidden + (size_t)(row0 + l15) * Hn + half * 2;
    const float* bRow0 = W + (size_t)l15        * Hn + half * 2;   // out cols 0..15
    const float* bRow1 = W + (size_t)(l15 + 16) * Hn + half * 2;   // out cols 16..31

    v8f c0 = {};
    v8f c1 = {};
    for (int h = 0; h < Hn; h += 4) {
        v2f a  = *(const v2f*)(aRow  + h);
        v2f b0 = *(const v2f*)(bRow0 + h);
        v2f b1 = *(const v2f*)(bRow1 + h);
        c0 = __builtin_amdgcn_wmma_f32_16x16x4_f32(false, a, false, b0,
                                                   (short)0, c0, false, false);
        c1 = __builtin_amdgcn_wmma_f32_16x16x4_f32(false, a, false, b1,
                                                   (short)0, c1, false, false);
    }

    // C/D f32 16x16 layout: col = lane&15, rows = vgpr + (lane>=16 ? 8 : 0)
    const int n     = l15;
    const int mbase = half * 8;
    float* out = feats + (size_t)(row0 + mbase) * Kn + n;
#pragma unroll
    for (int v = 0; v < 8; ++v) {
        out[(size_t)v * Kn]      = c0[v];
        out[(size_t)v * Kn + 16] = c1[v];
    }
}

// ---------------------------------------------------------------------------
// Kernel 2: per-chunk composite matrix in the log semiring.
// M_t[j,i] = trans[j,i] + feat_t[j];  chunk = M_{t0+L-1} ∘ ... ∘ M_{t0},
// (A∘B)[j,i] = logsumexp_k(A[j,k]+B[k,i]).  One thread per (j,i); ping-pong
// LDS buffers; lane index == i gives conflict-free LDS column reads.
// ---------------------------------------------------------------------------
__global__ __launch_bounds__(1024)
void crf_chunk_mats(const float* __restrict__ feats,
                    const float* __restrict__ trans,
                    float* __restrict__ chunkM) {
    __shared__ float sT[1024];
    __shared__ float sM[2][1024];
    const int tid = threadIdx.x;
    const int j = tid >> 5, i = tid & 31;
    const int c = blockIdx.x;
    const int t0 = c * CLEN;

    sT[tid] = trans[tid];
    __syncthreads();

    sM[0][tid] = sT[tid] + feats[(size_t)t0 * Kn + j];
    __syncthreads();

    int cur = 0;
    for (int t = t0 + 1; t < t0 + CLEN; ++t) {
        const float* Mi = sM[cur];
        const float fj = feats[(size_t)t * Kn + j];
        float x[32];
        float m = -3.4e38f;
#pragma unroll
        for (int k = 0; k < 32; ++k) {
            x[k] = sT[(j << 5) + k] + Mi[(k << 5) + i];
            m = fmaxf(m, x[k]);
        }
        float s = 0.0f;
#pragma unroll
        for (int k = 0; k < 32; ++k) s += __expf(x[k] - m);
        sM[cur ^ 1][tid] = fj + m + __logf(s);
        __syncthreads();
        cur ^= 1;
    }
    chunkM[(size_t)c * 1024 + tid] = sM[cur][tid];
}

// ---------------------------------------------------------------------------
// Kernel 3: gold path score, deterministic single-block tree reduction.
// ---------------------------------------------------------------------------
__global__ __launch_bounds__(1024)
void gold_reduce(const float* __restrict__ feats,
                 const float* __restrict__ trans,
                 const int* __restrict__ tags,
                 float* __restrict__ goldOut) {
    __shared__ float red[1024];
    const int tid = threadIdx.x;
    float acc = 0.0f;
    for (int nIdx = tid; nIdx < Tn; nIdx += 1024) {
        const int tg   = tags[nIdx];
        const int prev = (nIdx == 0) ? START : tags[nIdx - 1];
        acc += feats[(size_t)nIdx * Kn + tg] + trans[tg * Kn + prev];
    }
    red[tid] = acc;
    __syncthreads();
    for (int s = 512; s > 0; s >>= 1) {
        if (tid < s) red[tid] += red[tid + s];
        __syncthreads();
    }
    if (tid == 0) goldOut[0] = red[0] + trans[STOP * Kn + tags[Tn - 1]];
}

// ---------------------------------------------------------------------------
// Kernel 4: fold 512 chunk matrices into init vector, apply trans[STOP],
// subtract gold.  One wave; lane j owns fv[j].
// ---------------------------------------------------------------------------
__global__ __launch_bounds__(32)
void crf_finalize(const float* __restrict__ chunkM,
                  const float* __restrict__ trans,
                  const float* __restrict__ goldOut,
                  float* __restrict__ out) {
    __shared__ float sfv[32];
    const int lane = threadIdx.x;
    sfv[lane] = (lane == START) ? 0.0f : NEGV;
    __syncthreads();

    for (int c = 0; c < NCHUNK; ++c) {
        const float* M = chunkM + (size_t)c * 1024 + (size_t)lane * 32;
        float x[32];
        float m = -3.4e38f;
#pragma unroll
        for (int i = 0; i < 32; ++i) {
            x[i] = M[i] + sfv[i];
            m = fmaxf(m, x[i]);
        }
        float s = 0.0f;
#pragma unroll
        for (int i = 0; i < 32; ++i) s += __expf(x[i] - m);
        const float nv = m + __logf(s);
        __syncthreads();
        sfv[lane] = nv;
        __syncthreads();
    }

    if (lane == 0) {
        float m = -3.4e38f;
        for (int jj = 0; jj < 32; ++jj) m = fmaxf(m, sfv[jj] + trans[STOP * Kn + jj]);
        float s = 0.0f;
        for (int jj = 0; jj < 32; ++jj) s += __expf(sfv[jj] + trans[STOP * Kn + jj] - m);
        out[0] = (m + __logf(s)) - goldOut[0];
    }
}

// ---------------------------------------------------------------------------
extern "C" void kernel_launch(void* const* d_in, const int* in_sizes, int n_in,
                              void* d_out, int out_size, void* d_ws, size_t ws_size,
                              hipStream_t stream) {
    const float* hidden = (const float*)d_in[0];   // [T,B=1,H] fp32
    const float* W      = (const float*)d_in[1];   // [K,H]     fp32
    const float* trans  = (const float*)d_in[2];   // [K,K]     fp32
    const int*   tags   = (const int*)d_in[3];     // [T]       int32
    float* out = (float*)d_out;

    float* feats  = (float*)d_ws;                  // T*K  = 524288 floats (2 MB)
    float* chunkM = feats + (size_t)Tn * Kn;       // NCHUNK*1024 floats   (2 MB)
    float* gold   = chunkM + (size_t)NCHUNK * 1024;

    // 1) emission GEMM (WMMA f32): 1024 row-tiles, 8 waves/block
    feats_gemm_wmma<<<(Tn / 16) / 8, 256, 0, stream>>>(hidden, W, feats);

    // 2) per-chunk log-semiring composite matrices
    crf_chunk_mats<<<NCHUNK, 1024, 0, stream>>>(feats, trans, chunkM);

    // 3) gold path score
    gold_reduce<<<1, 1024, 0, stream>>>(feats, trans, tags, gold);

    // 4) fold chunks + final logsumexp − gold
    crf_finalize<<<1, 32, 0, stream>>>(chunkM, trans, gold, out);
}